// SigLoss_naive_2611340116457
// MI455X (gfx1250) — compile-verified
//
#include <hip/hip_runtime.h>
#include <hip/hip_bf16.h>

// Signature-kernel loss on MI455X (gfx1250).
//   X, Y : (A=128, M=256, D=16) fp32
//   loss = mean(kXX) + mean(kYY) - 2*mean(kXY)
//
// One workgroup (256 threads, 8 wave32) per (pair, batch) solve:
//   phase0: path differences dA, dB into LDS
//   phase1: inc = dA * dB^T via V_WMMA_F32_16X16X4_F32 into 255x256 LDS grid
//   phase2: antidiagonal wavefront PDE solve out of LDS (1 barrier / step)
// Then a single-block deterministic weighted reduction of the 384 partials.

typedef __attribute__((ext_vector_type(2))) float v2f;
typedef __attribute__((ext_vector_type(8))) float v8f;

#define AA   128
#define MM   255          // increments grid is 255x255
#define SEQ  256
#define DD   16
#define INC_STRIDE 256    // padded row stride for the LDS inc grid

__global__ __launch_bounds__(256) void sig_pde_kernel(const float* __restrict__ X,
                                                      const float* __restrict__ Y,
                                                      float* __restrict__ partials) {
    // 255*256 inc grid + 2 diff sequences + 3 rotating diagonals: ~297 KB LDS
    __shared__ float s_inc[MM * INC_STRIDE];   // 261120 B
    __shared__ float s_dA[MM * DD];            // 16320 B
    __shared__ float s_dB[MM * DD];            // 16320 B
    __shared__ float s_g0[256];
    __shared__ float s_g1[256];
    __shared__ float s_g2[256];

    const int tid = threadIdx.x;
    const int blk = blockIdx.x;
    const int a   = blk & (AA - 1);   // batch
    const int p   = blk >> 7;         // 0:XX 1:YY 2:XY

    const float* Aseq = (p == 1) ? Y : X;   // XX->X, YY->Y, XY->X
    const float* Bseq = (p == 0) ? X : Y;   // XX->X, YY->Y, XY->Y
    const long base = (long)a * SEQ * DD;

    // ---- phase 0: path increments along M ----
    for (int e = tid; e < MM * DD; e += 256) {
        const int i = e >> 4;       // 0..254
        const int k = e & 15;
        s_dA[e] = Aseq[base + (i + 1) * DD + k] - Aseq[base + i * DD + k];
        s_dB[e] = Bseq[base + (i + 1) * DD + k] - Bseq[base + i * DD + k];
    }
    // init wavefront diagonals (K boundary = 1)
    s_g0[tid] = 1.0f; s_g1[tid] = 1.0f; s_g2[tid] = 1.0f;
    __syncthreads();

    // ---- phase 1: inc tile grid via fp32 WMMA ----
    // 16x16 grid of 16x16 tiles; wave w handles tiles w, w+8, ...
    {
        const int wid  = tid >> 5;
        const int lane = tid & 31;
        const int h    = lane >> 4;   // half-wave select
        const int m    = lane & 15;

        for (int T = wid; T < 256; T += 8) {
            const int ti = T >> 4;
            const int tj = T & 15;
            int rowA = ti * 16 + m; if (rowA > MM - 1) rowA = MM - 1;  // clamp, discarded on store
            int colB = tj * 16 + m; if (colB > MM - 1) colB = MM - 1;

            v8f c = {};
#pragma unroll
            for (int kk = 0; kk < DD; kk += 4) {
                const int k0 = kk + 2 * h;
                v2f av, bv;
                // A-matrix 16x4 f32: lanes 0-15 row m K={kk,kk+1}; lanes 16-31 row m K={kk+2,kk+3}
                av.x = s_dA[rowA * DD + k0];
                av.y = s_dA[rowA * DD + k0 + 1];
                // B-matrix 4x16 f32: mirrored layout (row K striped across lanes)
                bv.x = s_dB[colB * DD + k0];
                bv.y = s_dB[colB * DD + k0 + 1];
                c = __builtin_amdgcn_wmma_f32_16x16x4_f32(
                        /*neg_a=*/false, av, /*neg_b=*/false, bv,
                        /*c_mod=*/(short)0, c, /*reuse_a=*/false, /*reuse_b=*/false);
            }
            // C/D 16x16 f32: VGPR r -> M=r (lanes 0-15) / M=r+8 (lanes 16-31), N=lane&15
#pragma unroll
            for (int r = 0; r < 8; ++r) {
                const int row = ti * 16 + r + 8 * h;
                const int col = tj * 16 + m;
                if (row < MM && col < MM) s_inc[row * INC_STRIDE + col] = c[r];
            }
        }
    }
    __syncthreads();

    // ---- phase 2: antidiagonal wavefront PDE solve ----
    // Thread t owns column j = t+1 of K (size 256x256, K[0][*]=K[*][0]=1).
    // K[i][j] = (K[i][j-1]+K[i-1][j])*(1 + g/2 + g^2/12) - K[i-1][j-1]*(1 - g^2/12)
    float* bD2 = s_g0;   // diagonal d-2
    float* bD1 = s_g1;   // diagonal d-1
    float* bW  = s_g2;   // write target (diagonal d)
    float kUp = 1.0f;    // K[i-1][j] carried along own column
    const int j = tid + 1;

    for (int d = 2; d <= 2 * MM; ++d) {
        const int i = d - j;
        if (tid < MM && i >= 1 && i <= MM) {
            const float kLeft = (tid == 0) ? 1.0f : bD1[tid - 1];
            const float kDiag = (tid == 0) ? 1.0f : bD2[tid - 1];
            const float g  = s_inc[(i - 1) * INC_STRIDE + tid];
            const float q  = g * g * (1.0f / 12.0f);
            const float c1 = 1.0f + 0.5f * g + q;
            const float c2 = 1.0f - q;
            const float v  = (kLeft + kUp) * c1 - kDiag * c2;
            bW[tid] = v;
            kUp = v;
        }
        __syncthreads();
        float* tmp = bD2; bD2 = bD1; bD1 = bW; bW = tmp;
    }

    // K[255][255] computed by thread 254 on the last diagonal
    if (tid == MM - 1) partials[blk] = kUp;
}

__global__ __launch_bounds__(256) void sig_reduce_kernel(const float* __restrict__ partials,
                                                         float* __restrict__ out) {
    __shared__ float s[256];
    const int t = threadIdx.x;
    float v = 0.0f;
    for (int idx = t; idx < 3 * AA; idx += 256) {
        const int p = idx >> 7;                                   // 0,1,2
        const float w = (p == 2) ? (-2.0f / (float)AA) : (1.0f / (float)AA);
        v += w * partials[idx];
    }
    s[t] = v;
    __syncthreads();
    for (int ofs = 128; ofs > 0; ofs >>= 1) {
        if (t < ofs) s[t] += s[t + ofs];
        __syncthreads();
    }
    if (t == 0) out[0] = s[0];
}

extern "C" void kernel_launch(void* const* d_in, const int* in_sizes, int n_in,
                              void* d_out, int out_size, void* d_ws, size_t ws_size,
                              hipStream_t stream) {
    (void)in_sizes; (void)n_in; (void)out_size; (void)ws_size;
    const float* X = (const float*)d_in[0];
    const float* Y = (const float*)d_in[1];
    float* out      = (float*)d_out;
    float* partials = (float*)d_ws;      // 384 floats = 1.5 KB of scratch

    sig_pde_kernel<<<dim3(3 * AA), dim3(256), 0, stream>>>(X, Y, partials);
    sig_reduce_kernel<<<dim3(1), dim3(256), 0, stream>>>(partials, out);
}